// SeqVAE_21852793602681
// MI455X (gfx1250) — compile-verified
//
#include <hip/hip_runtime.h>
#include <hip/hip_bf16.h>
#include <math.h>

#define B_   2048
#define S_   200
#define DX   64
#define ZD   32
#define H_   128
#define G_   384     /* 3H */
#define AE   32
#define TE   16
#define DIN  145
#define AV   100
#define TV   4

typedef __attribute__((ext_vector_type(2))) float v2f;
typedef __attribute__((ext_vector_type(8))) float v8f;

/* ---------------- setup kernels (tiny GEMMs, run once) ---------------- */

__global__ void k_xenc(const float* __restrict__ x, const float* __restrict__ w,
                       const float* __restrict__ b, float* __restrict__ xe) {
  int i = blockIdx.x * blockDim.x + threadIdx.x;      // over B*DX
  if (i >= B_ * DX) return;
  int bi = i / DX, d = i % DX;
  const float* xr = x + bi * DX;
  const float* wr = w + d * DX;
  float acc = b[d];
#pragma unroll 8
  for (int k = 0; k < DX; ++k) acc = fmaf(xr[k], wr[k], acc);
  xe[i] = acc;
}

__global__ void k_h0(const float* __restrict__ xe, const float* __restrict__ z,
                     const float* __restrict__ w, const float* __restrict__ b,
                     float* __restrict__ h) {
  int i = blockIdx.x * blockDim.x + threadIdx.x;      // over B*H
  if (i >= B_ * H_) return;
  int bi = i / H_, j = i % H_;
  const float* wr = w + j * (DX + ZD);
  float acc = b[j];
#pragma unroll 8
  for (int k = 0; k < DX; ++k) acc = fmaf(xe[bi * DX + k], wr[k], acc);
#pragma unroll 8
  for (int k = 0; k < ZD; ++k) acc = fmaf(z[bi * ZD + k], wr[DX + k], acc);
  h[i] = tanhf(acc);
}

/* gc[b,g] = x_enc . Wih[g,0:64] + z . Wih[g,112:144] + b_ih[g] */
__global__ void k_gc(const float* __restrict__ xe, const float* __restrict__ z,
                     const float* __restrict__ wih, const float* __restrict__ bih,
                     float* __restrict__ gc) {
  int i = blockIdx.x * blockDim.x + threadIdx.x;      // over B*G
  if (i >= B_ * G_) return;
  int bi = i / G_, g = i % G_;
  const float* wr = wih + g * DIN;
  float acc = bih[g];
#pragma unroll 8
  for (int k = 0; k < DX; ++k) acc = fmaf(xe[bi * DX + k], wr[k], acc);
#pragma unroll 8
  for (int k = 0; k < ZD; ++k) acc = fmaf(z[bi * ZD + k], wr[112 + k], acc);
  gc[i] = acc;
}

__global__ void k_aproj(const float* __restrict__ aemb, const float* __restrict__ wih,
                        float* __restrict__ ap) {
  int i = blockIdx.x * blockDim.x + threadIdx.x;      // over AV*G
  if (i >= AV * G_) return;
  int v = i / G_, g = i % G_;
  const float* wr = wih + g * DIN + 64;
  float acc = 0.f;
#pragma unroll 8
  for (int k = 0; k < AE; ++k) acc = fmaf(aemb[v * AE + k], wr[k], acc);
  ap[i] = acc;
}

__global__ void k_tproj(const float* __restrict__ temb, const float* __restrict__ wih,
                        float* __restrict__ tp) {
  int i = blockIdx.x * blockDim.x + threadIdx.x;      // over TV*G
  if (i >= TV * G_) return;
  int v = i / G_, g = i % G_;
  const float* wr = wih + g * DIN + 96;
  float acc = 0.f;
#pragma unroll
  for (int k = 0; k < TE; ++k) acc = fmaf(temb[v * TE + k], wr[k], acc);
  tp[i] = acc;
}

/* ---------------- per-timestep GRU kernel (WMMA fp32) ---------------- */
/* Block: 256 threads = 8 wave32s, owns 16 batch rows.
   Phase 1: gh = h_tile[16x128] @ Whh^T[128x384] via V_WMMA_F32_16X16X4_F32.
   Phase 2: fused gate math + hidden update.
   Phase 3: logits/prob heads. */
__global__ void __launch_bounds__(256, 1)
k_step(const float* __restrict__ whh, const float* __restrict__ bhh,
       const float* __restrict__ gc, const float* __restrict__ ap,
       const float* __restrict__ tp, const int* __restrict__ a,
       const int* __restrict__ t, const float* __restrict__ y,
       const float* __restrict__ mask, const float* __restrict__ wih,
       const float* __restrict__ yhw, const float* __restrict__ yhb,
       float* __restrict__ h, float* __restrict__ out, int s) {
  __shared__ float sh[16 * H_];        // h tile, reused for h_new
  __shared__ float sgh[16 * G_];       // gh tile
  __shared__ int   sa[16], st[16];
  __shared__ float spy[16], smk[16];

  const int tid = threadIdx.x;
  const int m0  = blockIdx.x * 16;

  for (int e = tid; e < 16 * H_; e += 256)
    sh[e] = h[(m0 + e / H_) * H_ + (e % H_)];
  if (tid < 16) {
    int b   = m0 + tid;
    sa[tid] = a[b * S_ + s];
    st[tid] = t[b * S_ + s];
    spy[tid] = (s == 0) ? 0.f : y[b * S_ + s - 1];
    smk[tid] = mask[b * S_ + s];
  }
  __syncthreads();

  /* -------- WMMA phase: 24 N-tiles of 16, 3 per wave, K=128 chained -------- */
  {
    const int w    = tid >> 5;
    const int lane = tid & 31;
    const int half = lane >> 4;      // K-half selector (A & B lane-half -> K)
    const int lm   = lane & 15;      // M for A, N for B/D
    for (int i = 0; i < 3; ++i) {
      const int n0 = (w * 3 + i) * 16;
      v8f c = (v8f){0.f, 0.f, 0.f, 0.f, 0.f, 0.f, 0.f, 0.f};
      const float* bp = whh + (n0 + lm) * H_;   // Whh row = B column, contiguous in K
#pragma unroll 8
      for (int k0 = 0; k0 < H_; k0 += 4) {
        const int kb = k0 + half * 2;
        v2f av; av.x = sh[lm * H_ + kb]; av.y = sh[lm * H_ + kb + 1];
        v2f bv = *(const v2f*)(bp + kb);        // 8B-aligned (kb even)
        c = __builtin_amdgcn_wmma_f32_16x16x4_f32(
                false, av, false, bv, (short)0, c, false, false);
      }
#pragma unroll
      for (int v = 0; v < 8; ++v)
        sgh[(half * 8 + v) * G_ + n0 + lm] = c[v];
    }
  }
  __syncthreads();

  /* -------- gate phase: r/u/n, hidden update -------- */
  for (int e = tid; e < 16 * H_; e += 256) {
    const int m = e / H_, j = e % H_;
    const int b = m0 + m;
    const int ai = sa[m], ti = st[m];
    const float py = spy[m];
    const float gr = gc[b * G_ + j] + ap[ai * G_ + j] + tp[ti * G_ + j]
                   + py * wih[j * DIN + 144] + sgh[m * G_ + j] + bhh[j];
    const float gu = gc[b * G_ + H_ + j] + ap[ai * G_ + H_ + j] + tp[ti * G_ + H_ + j]
                   + py * wih[(H_ + j) * DIN + 144] + sgh[m * G_ + H_ + j] + bhh[H_ + j];
    const float gin = gc[b * G_ + 2 * H_ + j] + ap[ai * G_ + 2 * H_ + j]
                    + tp[ti * G_ + 2 * H_ + j] + py * wih[(2 * H_ + j) * DIN + 144];
    const float ghn = sgh[m * G_ + 2 * H_ + j] + bhh[2 * H_ + j];
    const float r  = 1.f / (1.f + expf(-gr));
    const float u  = 1.f / (1.f + expf(-gu));
    const float n  = tanhf(gin + r * ghn);
    const float hn = (1.f - u) * n + u * sh[e];
    sh[e] = hn;                 // h_new for the head
    h[b * H_ + j] = hn;         // carry to next step
  }
  __syncthreads();

  /* -------- output head -------- */
  if (tid < 16) {
    const int m = tid, b = m0 + m;
    float acc = yhb[0];
#pragma unroll 8
    for (int j = 0; j < H_; ++j) acc = fmaf(sh[m * H_ + j], yhw[j], acc);
    const float logit = acc;
    const float prob  = 1.f / (1.f + expf(-logit));
    const float mk    = smk[m];
    out[b * S_ + s]           = mk * logit + (1.f - mk) * logit;  // == logit (fwd)
    out[B_ * S_ + b * S_ + s] = mk * prob  + (1.f - mk) * prob;   // == prob  (fwd)
  }
}

/* ---------------- launch ---------------- */

extern "C" void kernel_launch(void* const* d_in, const int* in_sizes, int n_in,
                              void* d_out, int out_size, void* d_ws, size_t ws_size,
                              hipStream_t stream) {
  const float* x      = (const float*)d_in[0];
  const int*   a      = (const int*)  d_in[1];
  const int*   t      = (const int*)  d_in[2];
  const float* y      = (const float*)d_in[3];
  const float* mask   = (const float*)d_in[4];
  const float* z      = (const float*)d_in[5];
  const float* xprojw = (const float*)d_in[6];
  const float* xprojb = (const float*)d_in[7];
  const float* aemb   = (const float*)d_in[8];
  const float* temb   = (const float*)d_in[9];
  const float* h0w    = (const float*)d_in[10];
  const float* h0b    = (const float*)d_in[11];
  const float* wih    = (const float*)d_in[12];
  const float* whh    = (const float*)d_in[13];
  const float* bih    = (const float*)d_in[14];
  const float* bhh    = (const float*)d_in[15];
  const float* yhw    = (const float*)d_in[16];
  const float* yhb    = (const float*)d_in[17];
  float* out = (float*)d_out;

  /* workspace layout (floats): xe | h | gc | A_proj | T_proj  (~4.9 MB) */
  float* xe = (float*)d_ws;
  float* h  = xe + (size_t)B_ * DX;
  float* gc = h  + (size_t)B_ * H_;
  float* ap = gc + (size_t)B_ * G_;
  float* tp = ap + (size_t)AV * G_;

  k_xenc <<<(B_ * DX + 255) / 256, 256, 0, stream>>>(x, xprojw, xprojb, xe);
  k_aproj<<<(AV * G_ + 255) / 256, 256, 0, stream>>>(aemb, wih, ap);
  k_tproj<<<(TV * G_ + 255) / 256, 256, 0, stream>>>(temb, wih, tp);
  k_h0   <<<(B_ * H_ + 255) / 256, 256, 0, stream>>>(xe, z, h0w, h0b, h);
  k_gc   <<<(B_ * G_ + 255) / 256, 256, 0, stream>>>(xe, z, wih, bih, gc);

  for (int s = 0; s < S_; ++s)
    k_step<<<B_ / 16, 256, 0, stream>>>(whh, bhh, gc, ap, tp, a, t, y, mask,
                                        wih, yhw, yhb, h, out, s);
}